// WpDecoder_5669356836354
// MI455X (gfx1250) — compile-verified
//
#include <hip/hip_runtime.h>
#include <math.h>

// ---------------------------------------------------------------------------
// MI455X (gfx1250) jerk-model scan.
// Pure streaming workload (~258 MB moved, ~0.24 GFLOP -> HBM-bound, ~11us
// floor at 23.3 TB/s). Strategy: Tensor Data Mover DMAs each block's input
// rows into LDS and the finished trajectory block back out (both regions are
// contiguous in global), while the recurrence runs in registers with hardware
// transcendentals. LDS traffic is 128-bit, global traffic is TDM bulk copies.
// ---------------------------------------------------------------------------

#define T_STEPS 30
#define BLK 64
#define DT_F 0.03f

typedef unsigned int v4u __attribute__((ext_vector_type(4)));
typedef int          v4i __attribute__((ext_vector_type(4)));
typedef int          v8i __attribute__((ext_vector_type(8)));

// Generic pointers to LDS carry the LDS byte offset in their low 32 bits
// (ISA: generic->LDS mapping truncates to addr[31:0]).
__device__ __forceinline__ unsigned lds_byte_off(const void* p) {
  return (unsigned)(unsigned long long)p;
}

__device__ __forceinline__ void tdm_wait0() {
#if __has_builtin(__builtin_amdgcn_s_wait_tensorcnt)
  __builtin_amdgcn_s_wait_tensorcnt(0);
#else
  asm volatile("s_wait_tensorcnt 0x0" ::: "memory");
#endif
}

// 2-D TDM transfer of dim1_rows rows x dim0_elts dwords (contiguous tensor,
// stride == dim0) between global memory and LDS.
__device__ __forceinline__ void tdm_copy_2d(unsigned lds_byte,
                                            unsigned long long gaddr,
                                            unsigned dim0_elts,
                                            unsigned dim1_rows,
                                            bool is_store) {
  // ---- D# group 0 (ISA 8.3): count=1 valid, lds_addr, global_addr, type=2
  v4u g0;
  g0[0] = 1u;
  g0[1] = lds_byte;
  g0[2] = (unsigned)(gaddr & 0xFFFFFFFFull);
  g0[3] = (unsigned)((gaddr >> 32) & 0x01FFFFFFull) | (2u << 30);

  // ---- D# group 1 (ISA 8.4): data_size=4B, dims/tile/stride, no multicast
  v8i g1 = {0, 0, 0, 0, 0, 0, 0, 0};
  g1[0] = (int)(2u << 16);                                          // data_size=2 (4B)
  g1[1] = (int)((dim0_elts & 0xFFFFu) << 16);                       // tensor_dim0[15:0]  @ bits 63:48
  g1[2] = (int)((dim0_elts >> 16) | ((dim1_rows & 0xFFFFu) << 16)); // tensor_dim0[31:16] | tensor_dim1[15:0]
  g1[3] = (int)((dim1_rows >> 16) | ((dim0_elts & 0xFFFFu) << 16)); // tensor_dim1[31:16] | tile_dim0
  g1[4] = (int)(dim1_rows & 0xFFFFu);                               // tile_dim1 (tile_dim2 = 0)
  g1[5] = (int)dim0_elts;                                           // tensor_dim0_stride[31:0]

  v4i g2 = {0, 0, 0, 0};
  v4i g3 = {0, 0, 0, 0};

#if __has_builtin(__builtin_amdgcn_tensor_load_to_lds)
#if defined(__clang_major__) && __clang_major__ >= 23
  v8i g4 = {0, 0, 0, 0, 0, 0, 0, 0};
  if (is_store)
    __builtin_amdgcn_tensor_store_from_lds(g0, g1, g2, g3, g4, 0);
  else
    __builtin_amdgcn_tensor_load_to_lds(g0, g1, g2, g3, g4, 0);
#else
  if (is_store)
    __builtin_amdgcn_tensor_store_from_lds(g0, g1, g2, g3, 0);
  else
    __builtin_amdgcn_tensor_load_to_lds(g0, g1, g2, g3, 0);
#endif
#else
  if (is_store)
    asm volatile("tensor_store_from_lds %0, %1, %2, %3"
                 :: "s"(g0), "s"(g1), "s"(g2), "s"(g3) : "memory");
  else
    asm volatile("tensor_load_to_lds %0, %1, %2, %3"
                 :: "s"(g0), "s"(g1), "s"(g2), "s"(g3) : "memory");
#endif
}

// One step of the jerk model (matches reference semantics & op order).
__device__ __forceinline__ void jerk_step(float c0, float c1,
                                          float& x, float& y, float& psi,
                                          float& v, float& pedal, float& steer) {
  float jerk    = fminf(fmaxf(c0 * 4.0f, -4.0f), 4.0f);
  float srate   = fminf(fmaxf(c1 * 0.5f, -0.5f), 0.5f);
  float pedal_n = fminf(fmaxf(pedal + jerk * DT_F, -5.0f), 5.0f);
  float steer_n = fminf(fmaxf(steer + srate * DT_F, -0.5f), 0.5f);
  float v_n     = fminf(fmaxf(v + pedal_n * DT_F, 0.0f), 10.0f);
  float psi_dot = fminf(fmaxf(v * __tanf(steer_n) / 2.5f, -1.57f), 1.57f); // old v
  float psi_n   = psi_dot * DT_F + psi;
  float x_n     = v_n * __cosf(psi_n) * DT_F + x;
  float y_n     = v_n * __sinf(psi_n) * DT_F + y;
  x = x_n; y = y_n; psi = psi_n; v = v_n; pedal = pedal_n; steer = steer_n;
}

__global__ __launch_bounds__(BLK) void jerk_scan_tdm(const float* __restrict__ z,
                                                     const float* __restrict__ init_state,
                                                     float* __restrict__ out) {
  __shared__ __align__(16) float s_in[BLK * 2 * T_STEPS];   // 64 x 240B = 15 KB
  __shared__ __align__(16) float s_out[BLK * 6 * T_STEPS];  // 64 x 720B = 45 KB

  const int tid = (int)threadIdx.x;
  const long long b0 = (long long)blockIdx.x * BLK;
  const long long b  = b0 + tid;

  // Issue the per-thread init-state loads FIRST so they overlap the TDM fill
  // and the workgroup barrier (6 floats per row, 8B-aligned -> three b64).
  const float* st = init_state + b * 6;
  float2 p01 = *(const float2*)(st + 0);
  float2 p23 = *(const float2*)(st + 2);
  float2 p45 = *(const float2*)(st + 4);

  // Wave 0 DMAs this block's control rows (contiguous in global) into LDS.
  if (tid < 32) {
    tdm_copy_2d(lds_byte_off(s_in),
                (unsigned long long)(z + b0 * (2 * T_STEPS)),
                2 * T_STEPS, BLK, /*is_store=*/false);
    tdm_wait0();
  }
  __syncthreads();

  float x = p01.x, y = p01.y, psi = p23.x, v = p23.y, pedal = p45.x, steer = p45.y;

  const float* row  = s_in  + tid * (2 * T_STEPS);   // 240B-aligned
  float*       orow = s_out + tid * (6 * T_STEPS);   // 720B-aligned

  // Two steps per iteration: one ds_load_b128 (4 controls), three
  // ds_store_b128 (12 state floats). All LDS accesses 16B-aligned.
  for (int u = 0; u < T_STEPS / 2; ++u) {
    float4 c = *(const float4*)(row + 4 * u);
    jerk_step(c.x, c.y, x, y, psi, v, pedal, steer);
    float4 o0 = {x, y, psi, v};
    float p0 = pedal, s0 = steer;
    jerk_step(c.z, c.w, x, y, psi, v, pedal, steer);
    float4 o1 = {p0, s0, x, y};
    float4 o2 = {psi, v, pedal, steer};
    *(float4*)(orow + 12 * u + 0) = o0;
    *(float4*)(orow + 12 * u + 4) = o1;
    *(float4*)(orow + 12 * u + 8) = o2;
  }

  __syncthreads();

  // Whole-block trajectory is a contiguous 46,080B global region: DMA it out.
  if (tid < 32) {
    tdm_copy_2d(lds_byte_off(s_out),
                (unsigned long long)(out + b0 * (6 * T_STEPS)),
                6 * T_STEPS, BLK, /*is_store=*/true);
    tdm_wait0();  // hold LDS alive until the TDM has drained it
  }
}

// Plain tail kernel for B % BLK rows (generic safety net).
__global__ void jerk_scan_tail(const float* __restrict__ z,
                               const float* __restrict__ init_state,
                               float* __restrict__ out, int b_start, int B) {
  int b = b_start + (int)(blockIdx.x * blockDim.x + threadIdx.x);
  if (b >= B) return;
  const float* st = init_state + (long long)b * 6;
  float x = st[0], y = st[1], psi = st[2], v = st[3], pedal = st[4], steer = st[5];
  const float* row  = z   + (long long)b * (2 * T_STEPS);
  float*       orow = out + (long long)b * (6 * T_STEPS);
  for (int t = 0; t < T_STEPS; ++t) {
    jerk_step(row[2 * t], row[2 * t + 1], x, y, psi, v, pedal, steer);
    orow[6 * t + 0] = x;     orow[6 * t + 1] = y;
    orow[6 * t + 2] = psi;   orow[6 * t + 3] = v;
    orow[6 * t + 4] = pedal; orow[6 * t + 5] = steer;
  }
}

extern "C" void kernel_launch(void* const* d_in, const int* in_sizes, int n_in,
                              void* d_out, int out_size, void* d_ws, size_t ws_size,
                              hipStream_t stream) {
  const float* z          = (const float*)d_in[0];
  const float* init_state = (const float*)d_in[1];
  float* out = (float*)d_out;

  const int B  = in_sizes[1] / 6;   // init_state is [B, 6]
  const int nb = B / BLK;
  if (nb > 0)
    jerk_scan_tdm<<<nb, BLK, 0, stream>>>(z, init_state, out);
  const int rem = B - nb * BLK;
  if (rem > 0)
    jerk_scan_tail<<<(rem + 63) / 64, 64, 0, stream>>>(z, init_state, out,
                                                       nb * BLK, B);
}